// GraphConv_35708358099684
// MI455X (gfx1250) — compile-verified
//
#include <hip/hip_runtime.h>
#include <stdint.h>

// ---------------------------------------------------------------------------
// SGL / LightGCN propagation for MI455X (gfx1250, wave32).
//   hop k: next = SpMM(adj, ego)  (COO scatter-add, L2-resident tables)
//          ego  = next + sign(next) * (u / ||u||_row) * EPS
//   outputs: mean over the 3 hop results, plus the hop-0 result.
//
// Roofline: ~1.84 GFLOP vs ~0.4 GB moved -> memory/atomic bound; WMMA is
// inapplicable (unsorted COO). CDNA5 paths: async global->LDS edge staging
// (ASYNCcnt, double buffered), v_readlane scalarized edge broadcast, f32
// global atomics resolved in the 192MB L2, NT-hint stores for write-once
// outputs so they don't evict the resident tables.
// ---------------------------------------------------------------------------

#define N_USERS 100000
#define N_ITEMS 50000
#define N_NODES 150000
#define NNZ     4800000
#define DIM     64
#define EPSV    0.1f

#define WPB   8                        // waves per block (256 threads)
#define CHUNK 32                       // edges per wave per chunk
#define STAGE_BYTES (3 * CHUNK * 4)    // rows|cols|vals per buffer = 384 B

typedef float v2f __attribute__((ext_vector_type(2)));   // native vector for NT stores

// ---- async global -> LDS (gfx1250). vdst = LDS byte offset, addr = 64-bit
// global address. Tracked with ASYNCcnt. ----
__device__ __forceinline__ void async_ld_b32(uint32_t lds_off, const void* g) {
  uint64_t ga = (uint64_t)g;
  asm volatile("global_load_async_to_lds_b32 %0, %1, off"
               :: "v"(lds_off), "v"(ga) : "memory");
}

// ---- cheap device-side uniform hash (stands in for jax threefry; noise
// structure — per-row L2 normalization, sign(x)*u*EPS — is preserved) ----
__device__ __forceinline__ float uhash(uint32_t x) {
  x ^= x >> 17; x *= 0xed5ad4bbu;
  x ^= x >> 11; x *= 0xac4c1b51u;
  x ^= x >> 15; x *= 0x31848babu;
  x ^= x >> 14;
  return ((float)(x >> 8) + 0.5f) * (1.0f / 16777216.0f);
}

// ---------------------------------------------------------------------------
__global__ __launch_bounds__(256) void concat_embed(
    const float* __restrict__ u, const float* __restrict__ it,
    float4* __restrict__ ego4) {
  size_t i = (size_t)blockIdx.x * blockDim.x + threadIdx.x;
  const size_t nu4 = (size_t)N_USERS * DIM / 4;
  const size_t ni4 = (size_t)N_ITEMS * DIM / 4;
  if (i < nu4)            ego4[i] = ((const float4*)u)[i];
  else if (i < nu4 + ni4) ego4[i] = ((const float4*)it)[i - nu4];
}

__global__ __launch_bounds__(256) void zero_f4(float4* __restrict__ p, size_t n4) {
  size_t i = (size_t)blockIdx.x * blockDim.x + threadIdx.x;
  if (i < n4) p[i] = make_float4(0.f, 0.f, 0.f, 0.f);
}

// ---------------------------------------------------------------------------
// SpMM: y[row] += val * x[col].  One wave per 32-edge chunk; lane owns dims
// {2*lane, 2*lane+1} so each row gather is a single coalesced b64 per lane
// (256 B per wave per edge). Edge triples are staged into LDS with async
// loads, double-buffered so the next chunk streams in while the current
// chunk's gathers/atomics are in flight.
// ---------------------------------------------------------------------------
__global__ __launch_bounds__(256) void spmm_atomic(
    const int*   __restrict__ rows,
    const int*   __restrict__ cols,
    const float* __restrict__ vals,
    const float* __restrict__ x,
    float*       __restrict__ y,
    int nchunks, int nwaves) {
  __shared__ int s_stage[WPB][2][3 * CHUNK];   // [wave][buf][rows|cols|vals]

  const int lane = threadIdx.x & 31;
  const int wib  = threadIdx.x >> 5;
  const int gw   = blockIdx.x * WPB + wib;

  // Flat shared address truncates to the LDS byte offset (aperture rule).
  const uint32_t lds_wave = (uint32_t)(uintptr_t)(&s_stage[wib][0][0]);

  int chunk = gw;
  int buf = 0;
  if (chunk < nchunks) {
    const size_t e = (size_t)chunk * CHUNK + lane;
    async_ld_b32(lds_wave + 0   + lane * 4, rows + e);
    async_ld_b32(lds_wave + 128 + lane * 4, cols + e);
    async_ld_b32(lds_wave + 256 + lane * 4, vals + e);
  }

  while (chunk < nchunks) {
    const int nextc = chunk + nwaves;
    if (nextc < nchunks) {
      const uint32_t nb = lds_wave + (uint32_t)(buf ^ 1) * STAGE_BYTES;
      const size_t e = (size_t)nextc * CHUNK + lane;
      async_ld_b32(nb + 0   + lane * 4, rows + e);
      async_ld_b32(nb + 128 + lane * 4, cols + e);
      async_ld_b32(nb + 256 + lane * 4, vals + e);
      __builtin_amdgcn_s_wait_asynccnt(3);   // current buffer has landed
    } else {
      __builtin_amdgcn_s_wait_asynccnt(0);
    }
    asm volatile("" ::: "memory");

    // each lane grabs one edge triple, then scalar-broadcast via v_readlane
    const int my_r = s_stage[wib][buf][lane];
    const int my_c = s_stage[wib][buf][32 + lane];
    const int my_v = s_stage[wib][buf][64 + lane];

    for (int j = 0; j < CHUNK; ++j) {
      const int   r = __builtin_amdgcn_readlane(my_r, j);
      const int   c = __builtin_amdgcn_readlane(my_c, j);
      const float v = __int_as_float(__builtin_amdgcn_readlane(my_v, j));
      const v2f xv = ((const v2f*)(x + (size_t)c * DIM))[lane];  // b64 gather
      float* yd = y + (size_t)r * DIM + 2 * lane;
      atomicAdd(yd + 0, v * xv.x);
      atomicAdd(yd + 1, v * xv.y);
    }

    buf ^= 1;
    chunk = nextc;
  }
}

// ---------------------------------------------------------------------------
// Perturb + accumulate + emit (one wave per node, lane owns dims
// {2*lane, 2*lane+1}; all b64 accesses). Fuses:
//   ego = h + sign(h)*u*EPS/||u||;  acc += ego;  layer0/mean outputs (NT);
//   and re-zeros h for the next hop's scatter (saves a sweep kernel).
// ---------------------------------------------------------------------------
__global__ __launch_bounds__(256) void perturb_update(
    float*       __restrict__ h,
    float*       __restrict__ ego,
    float*       __restrict__ acc,
    float*       __restrict__ out_mean,
    float*       __restrict__ out_layer,
    int hop) {
  const int node = blockIdx.x * WPB + (threadIdx.x >> 5);
  const int lane = threadIdx.x & 31;
  if (node >= N_NODES) return;

  const uint32_t seed = (uint32_t)hop * 0x9e3779b9u + (uint32_t)node * 64u;
  const float u0 = uhash(seed + 2u * (uint32_t)lane);
  const float u1 = uhash(seed + 2u * (uint32_t)lane + 1u);

  float ss = u0 * u0 + u1 * u1;
  #pragma unroll
  for (int off = 16; off > 0; off >>= 1) ss += __shfl_xor(ss, off, 32);
  const float inv = rsqrtf(ss) * EPSV;       // EPS / ||u||

  const size_t i2 = (size_t)node * (DIM / 2) + lane;    // v2f index
  v2f* h2   = (v2f*)h;
  v2f* ego2 = (v2f*)ego;
  v2f* acc2 = (v2f*)acc;

  const v2f xv = h2[i2];
  if (hop < 2) h2[i2] = (v2f){0.f, 0.f};                // pre-zero next scatter

  const float s0 = (xv.x > 0.f) ? 1.f : ((xv.x < 0.f) ? -1.f : 0.f);
  const float s1 = (xv.y > 0.f) ? 1.f : ((xv.y < 0.f) ? -1.f : 0.f);
  const v2f yv = (v2f){fmaf(s0 * u0, inv, xv.x),
                       fmaf(s1 * u1, inv, xv.y)};
  ego2[i2] = yv;

  v2f av;
  if (hop == 0) {
    av = yv;
    __builtin_nontemporal_store(yv, &((v2f*)out_layer)[i2]);   // NT: write-once
  } else {
    const v2f p = acc2[i2];
    av = p + yv;
  }
  if (hop == 2) {
    const v2f m = av * (1.0f / 3.0f);
    __builtin_nontemporal_store(m, &((v2f*)out_mean)[i2]);     // NT: write-once
  } else {
    acc2[i2] = av;
  }
}

// ---------------------------------------------------------------------------
extern "C" void kernel_launch(void* const* d_in, const int* in_sizes, int n_in,
                              void* d_out, int out_size, void* d_ws, size_t ws_size,
                              hipStream_t stream) {
  const float* user_embed = (const float*)d_in[0];
  const float* item_embed = (const float*)d_in[1];
  const int*   adj_rows   = (const int*)d_in[2];
  const int*   adj_cols   = (const int*)d_in[3];
  const float* adj_vals   = (const float*)d_in[4];

  float* out_mean  = (float*)d_out;
  float* out_layer = out_mean + (size_t)N_NODES * DIM;

  const size_t tblN = (size_t)N_NODES * DIM;   // 9.6M floats = 38.4 MB
  float* ego = (float*)d_ws;
  float* nxt = ego + tblN;
  float* acc = nxt + tblN;
  (void)ws_size; (void)n_in; (void)in_sizes; (void)out_size;

  const size_t tot4 = tblN / 4;                          // 2.4M float4
  const int cpBlocks = (int)((tot4 + 255) / 256);

  concat_embed<<<cpBlocks, 256, 0, stream>>>(user_embed, item_embed, (float4*)ego);
  zero_f4<<<cpBlocks, 256, 0, stream>>>((float4*)nxt, tot4);   // hop-0 scatter target

  const int nchunks    = NNZ / CHUNK;                    // 150000, exact
  const int spmmBlocks = 1875;                           // 15000 waves -> 10 chunks each
  const int nwaves     = spmmBlocks * WPB;
  const int nodeBlocks = (N_NODES + WPB - 1) / WPB;      // 18750

  for (int hop = 0; hop < 3; ++hop) {
    spmm_atomic<<<spmmBlocks, 256, 0, stream>>>(adj_rows, adj_cols, adj_vals,
                                                ego, nxt, nchunks, nwaves);
    // also re-zeros nxt (hops 0,1) for the next hop's scatter
    perturb_update<<<nodeBlocks, 256, 0, stream>>>(nxt, ego, acc,
                                                   out_mean, out_layer, hop);
  }
}